// GC_meanpool_39917426049438
// MI455X (gfx1250) — compile-verified
//
#include <hip/hip_runtime.h>
#include <hip/hip_bf16.h>

// ---------------------------------------------------------------------------
// GraphConv (PyG) x3 + global_mean_pool + MLP head + log_softmax for gfx1250.
// N=40000, E=640000, H=128, C=10, G=256, L=3.
//  - Edge aggregation: device-built CSR + register gather (no f32 atomics in
//    the hot loop), next-row global_prefetch.
//  - GEMMs: V_WMMA_F32_16X16X4_F32, weights staged in 128.5 KB LDS (CDNA5
//    320KB/WGP), swizzled for ds_load_b64 B-fragments, A read exactly once.
//    Staging vectorized: float4 global reads, paired LDS stores.
// ---------------------------------------------------------------------------

#define GCN_N 40000
#define GCN_E 640000
#define GCN_H 128
#define GCN_C 10
#define GCN_G 256
#define GCN_L 3

#define NTILES (GCN_N / 16)          // 2500 row tiles
#define KPAIR_STRIDE 257             // floats per k-pair row in LDS (bank skew)
#define LDS_MAT_FLOATS (64 * KPAIR_STRIDE)      // 16448 floats per matrix
#define LDS_TOTAL_FLOATS (2 * LDS_MAT_FLOATS)   // 32896 floats = 131584 B

typedef float v2f __attribute__((ext_vector_type(2)));
typedef float v8f __attribute__((ext_vector_type(8)));

// ---------------- generic fills --------------------------------------------
__global__ void gcn_zero_f4(float4* __restrict__ p, int n4) {
    int i = blockIdx.x * blockDim.x + threadIdx.x;
    if (i < n4) p[i] = make_float4(0.f, 0.f, 0.f, 0.f);
}

__global__ void gcn_zero_i(int* __restrict__ p, int n) {
    int i = blockIdx.x * blockDim.x + threadIdx.x;
    if (i < n) p[i] = 0;
}

__global__ void gcn_copy_i(const int* __restrict__ a, int* __restrict__ b, int n) {
    int i = blockIdx.x * blockDim.x + threadIdx.x;
    if (i < n) b[i] = a[i];
}

// ---------------- per-graph node counts ------------------------------------
__global__ void gcn_count(const int* __restrict__ batch, float* __restrict__ counts, int n) {
    int i = blockIdx.x * blockDim.x + threadIdx.x;
    if (i < n) atomicAdd(&counts[batch[i]], 1.0f);
}

__global__ void gcn_inv(const float* __restrict__ counts, float* __restrict__ inv, int g) {
    int i = blockIdx.x * blockDim.x + threadIdx.x;
    if (i < g) {
        float c = counts[i];
        inv[i] = (c > 0.f) ? (1.0f / c) : 0.f;
    }
}

__global__ void gcn_node_scale(const int* __restrict__ batch, const float* __restrict__ inv,
                               float* __restrict__ nscale, int n) {
    int i = blockIdx.x * blockDim.x + threadIdx.x;
    if (i < n) nscale[i] = inv[batch[i]];
}

// ---------------- CSR build: degree -> scan -> fill ------------------------
__global__ void gcn_deg(const int* __restrict__ dst, int* __restrict__ deg, int e) {
    int i = blockIdx.x * blockDim.x + threadIdx.x;
    if (i < e) atomicAdd(&deg[dst[i]], 1);
}

#define SCAN_B 256
__global__ void gcn_chunk_reduce(const int* __restrict__ deg, int* __restrict__ chunkSum, int n) {
    __shared__ int s[SCAN_B];
    int i = blockIdx.x * SCAN_B + threadIdx.x;
    s[threadIdx.x] = (i < n) ? deg[i] : 0;
    __syncthreads();
#pragma unroll
    for (int o = SCAN_B / 2; o > 0; o >>= 1) {
        if (threadIdx.x < o) s[threadIdx.x] += s[threadIdx.x + o];
        __syncthreads();
    }
    if (threadIdx.x == 0) chunkSum[blockIdx.x] = s[0];
}

__global__ void gcn_chunk_scan(const int* __restrict__ chunkSum, int* __restrict__ chunkOff,
                               int nchunks) {
    if (threadIdx.x == 0 && blockIdx.x == 0) {
        int acc = 0;
        for (int i = 0; i < nchunks; ++i) { chunkOff[i] = acc; acc += chunkSum[i]; }
    }
}

__global__ void gcn_chunk_apply(const int* __restrict__ deg, const int* __restrict__ chunkOff,
                                int* __restrict__ rowptr, int n) {
    __shared__ int s[SCAN_B];
    int i = blockIdx.x * SCAN_B + threadIdx.x;
    int v = (i < n) ? deg[i] : 0;
    s[threadIdx.x] = v;
    __syncthreads();
#pragma unroll
    for (int o = 1; o < SCAN_B; o <<= 1) {
        int t = (threadIdx.x >= o) ? s[threadIdx.x - o] : 0;
        __syncthreads();
        s[threadIdx.x] += t;
        __syncthreads();
    }
    if (i < n) rowptr[i] = chunkOff[blockIdx.x] + s[threadIdx.x] - v;  // exclusive
}

__global__ void gcn_csr_fill(const int* __restrict__ src, const int* __restrict__ dst,
                             int* __restrict__ pos, int* __restrict__ csr_src, int e) {
    int i = blockIdx.x * blockDim.x + threadIdx.x;
    if (i < e) {
        int p = atomicAdd(&pos[dst[i]], 1);
        csr_src[p] = src[i];
    }
}

// ---------------- CSR gather: agg[n] = sum_{j in N(n)} h[j] ----------------
// one wave per destination node; lane covers 4 contiguous features.
__global__ __launch_bounds__(256)
void gcn_gather(const float* __restrict__ h, const int* __restrict__ rowptr,
                const int* __restrict__ csr_src, float* __restrict__ agg, int nnodes) {
    int wid = (blockIdx.x * blockDim.x + threadIdx.x) >> 5;
    if (wid >= nnodes) return;
    int f = (threadIdx.x & 31) * 4;
    int beg = rowptr[wid];
    int end = (wid + 1 < nnodes) ? rowptr[wid + 1] : GCN_E;
    float ax = 0.f, ay = 0.f, az = 0.f, aw = 0.f;
    if (beg < end) {
        int s = csr_src[beg];
        for (int j = beg; j < end; ++j) {
            int snext = (j + 1 < end) ? csr_src[j + 1] : s;
            __builtin_prefetch(h + (size_t)snext * GCN_H + f, 0, 1);
            const float4 v = *(const float4*)(h + (size_t)s * GCN_H + f);
            ax += v.x; ay += v.y; az += v.z; aw += v.w;
            s = snext;
        }
    }
    *(float4*)(agg + (size_t)wid * GCN_H + f) = make_float4(ax, ay, az, aw);
}

// ---------------- fused dual GEMM via WMMA f32 16x16x4 ---------------------
// out[r,c] = (agg@wr + h@wroot + br)[r,c] * nscale[r]
// block = 256 threads = 8 waves; wave w owns row tile (blockIdx*8 + w) of 16
// rows x all 128 cols (8 x v8f accumulators). Both weight matrices staged in
// LDS, swizzled so each lane's (K, K+1) B pair is one ds_load_b64:
//   ldsW[(k>>1)*257 + col*2 + (k&1)]
__global__ __launch_bounds__(256)
void gcn_gemm_wmma(const float* __restrict__ agg, const float* __restrict__ h,
                   const float* __restrict__ wr, const float* __restrict__ wroot,
                   const float* __restrict__ br, const float* __restrict__ nscale,
                   float* __restrict__ out, int ntiles) {
    extern __shared__ float lds[];
    float* ldsR = lds;
    float* ldsT = lds + LDS_MAT_FLOATS;

    const int tid  = threadIdx.x;
    const int wave = tid >> 5;
    const int lane = tid & 31;
    const int half = lane >> 4;   // 0: K pair {kk,kk+1}; 1: K pair {kk+2,kk+3}
    const int l15  = lane & 15;

    // stage weights: float4 coalesced global reads -> swizzled LDS stores.
    // element (k,c) -> lds[(k>>1)*257 + 2c + (k&1)]; a float4 covers c..c+3
    // of one row k, i.e. LDS offsets 2c, 2c+2, 2c+4, 2c+6 (stride 2).
    for (int i4 = tid; i4 < (GCN_H * GCN_H) / 4; i4 += 256) {
        int k  = i4 >> 5;               // row (128 columns = 32 float4s)
        int c0 = (i4 & 31) << 2;        // first column of this float4
        int li = (k >> 1) * KPAIR_STRIDE + (c0 << 1) + (k & 1);
        const float4 vR = *(const float4*)(wr + (size_t)i4 * 4);
        const float4 vT = *(const float4*)(wroot + (size_t)i4 * 4);
        ldsR[li + 0] = vR.x; ldsR[li + 2] = vR.y;
        ldsR[li + 4] = vR.z; ldsR[li + 6] = vR.w;
        ldsT[li + 0] = vT.x; ldsT[li + 2] = vT.y;
        ldsT[li + 4] = vT.z; ldsT[li + 6] = vT.w;
    }
    __syncthreads();

    int tile  = blockIdx.x * 8 + wave;
    int tileC = (tile < ntiles) ? tile : (ntiles - 1);   // clamp: EXEC stays full
    int r0    = tileC * 16;

    const float* aAgg = agg + (size_t)(r0 + l15) * GCN_H + 2 * half;
    const float* aH   = h   + (size_t)(r0 + l15) * GCN_H + 2 * half;

    v8f acc[8] = {};
#pragma unroll 1
    for (int kk = 0; kk < GCN_H; kk += 4) {
        v2f a0 = *(const v2f*)(aAgg + kk);
        v2f a1 = *(const v2f*)(aH + kk);
        int kb = (kk >> 1) + half;                       // k-pair row for this lane
        const float* pR = ldsR + kb * KPAIR_STRIDE + (l15 << 1);
        const float* pT = ldsT + kb * KPAIR_STRIDE + (l15 << 1);
#pragma unroll
        for (int c = 0; c < 8; ++c) {
            v2f b0 = *(const v2f*)(pR + c * 32);
            acc[c] = __builtin_amdgcn_wmma_f32_16x16x4_f32(false, a0, false, b0,
                                                           (short)0, acc[c], false, false);
        }
#pragma unroll
        for (int c = 0; c < 8; ++c) {
            v2f b1 = *(const v2f*)(pT + c * 32);
            acc[c] = __builtin_amdgcn_wmma_f32_16x16x4_f32(false, a1, false, b1,
                                                           (short)0, acc[c], false, false);
        }
    }

    if (tile < ntiles) {
#pragma unroll
        for (int c = 0; c < 8; ++c) {
            int col = c * 16 + l15;
            float bias = br[col];
#pragma unroll
            for (int i = 0; i < 8; ++i) {
                int row = r0 + i + 8 * half;             // C/D: VGPR i -> M=i (+8 hi half)
                out[(size_t)row * GCN_H + col] = (acc[c][i] + bias) * nscale[row];
            }
        }
    }
}

// ---------------- global mean pool (sum part) ------------------------------
__global__ __launch_bounds__(256)
void gcn_pool(const float* __restrict__ x, const int* __restrict__ batch,
              float* __restrict__ pooled, int nnodes) {
    int t = blockIdx.x * blockDim.x + threadIdx.x;
    int n = t >> 5;
    if (n >= nnodes) return;
    int f = (t & 31) * 4;
    int g = batch[n];
    const float4 v = *(const float4*)(x + (size_t)n * GCN_H + f);
    float* p = pooled + (size_t)g * GCN_H + f;
    atomicAdd(p + 0, v.x);
    atomicAdd(p + 1, v.y);
    atomicAdd(p + 2, v.z);
    atomicAdd(p + 3, v.w);
}

// ---------------- MLP head + log_softmax (per graph) -----------------------
__global__ __launch_bounds__(128)
void gcn_head(const float* __restrict__ pooled, const float* __restrict__ inv,
              const float* __restrict__ w1, const float* __restrict__ b1,
              const float* __restrict__ w2, const float* __restrict__ b2,
              float* __restrict__ out) {
    __shared__ float sp[GCN_H];
    __shared__ float sh[GCN_H];
    __shared__ float slog[16];
    const int g = blockIdx.x;
    const int t = threadIdx.x;

    sp[t] = pooled[(size_t)g * GCN_H + t] * inv[g];
    __syncthreads();

    float acc = b1[t];
#pragma unroll 4
    for (int k = 0; k < GCN_H; ++k) acc += sp[k] * w1[(size_t)k * GCN_H + t];
    sh[t] = fmaxf(acc, 0.f);
    __syncthreads();

    if (t < GCN_C) {
        float a = b2[t];
#pragma unroll 4
        for (int k = 0; k < GCN_H; ++k) a += sh[k] * w2[(size_t)k * GCN_C + t];
        slog[t] = a;
    }
    __syncthreads();

    if (t == 0) {
        float m = slog[0];
        for (int c = 1; c < GCN_C; ++c) m = fmaxf(m, slog[c]);
        float s = 0.f;
        for (int c = 0; c < GCN_C; ++c) s += expf(slog[c] - m);
        float lse = m + logf(s);
        for (int c = 0; c < GCN_C; ++c) out[(size_t)g * GCN_C + c] = slog[c] - lse;
    }
}

// ---------------------------------------------------------------------------
extern "C" void kernel_launch(void* const* d_in, const int* in_sizes, int n_in,
                              void* d_out, int out_size, void* d_ws, size_t ws_size,
                              hipStream_t stream) {
    const float* x      = (const float*)d_in[0];
    const int*   ei     = (const int*)d_in[1];
    const int*   batch  = (const int*)d_in[2];
    const float* w_rel  = (const float*)d_in[3];
    const float* b_rel  = (const float*)d_in[4];
    const float* w_root = (const float*)d_in[5];
    const float* w1     = (const float*)d_in[6];
    const float* b1     = (const float*)d_in[7];
    const float* w2     = (const float*)d_in[8];
    const float* b2     = (const float*)d_in[9];
    float* out = (float*)d_out;

    const int* src = ei;
    const int* dst = ei + GCN_E;

    const int nchunks = (GCN_N + SCAN_B - 1) / SCAN_B;   // 157

    // ---- workspace carve-up ----
    float* ws     = (float*)d_ws;
    float* xA     = ws;                          // N*H
    float* xB     = xA + (size_t)GCN_N * GCN_H;  // N*H
    float* agg    = xB + (size_t)GCN_N * GCN_H;  // N*H
    float* counts = agg + (size_t)GCN_N * GCN_H; // G
    float* inv    = counts + GCN_G;              // G
    float* nscale = inv + GCN_G;                 // N
    float* pooled = nscale + GCN_N;              // G*H
    int* ib       = (int*)(pooled + (size_t)GCN_G * GCN_H);
    int* deg      = ib;                          // N
    int* rowptr   = deg + GCN_N;                 // N
    int* pos      = rowptr + GCN_N;              // N
    int* chunkSum = pos + GCN_N;                 // nchunks
    int* chunkOff = chunkSum + nchunks;          // nchunks
    int* csr_src  = chunkOff + nchunks;          // E

    // ---- graph stats ----
    gcn_zero_f4<<<1, 64, 0, stream>>>((float4*)counts, GCN_G / 4);
    gcn_count<<<(GCN_N + 255) / 256, 256, 0, stream>>>(batch, counts, GCN_N);
    gcn_inv<<<1, GCN_G, 0, stream>>>(counts, inv, GCN_G);
    gcn_node_scale<<<(GCN_N + 255) / 256, 256, 0, stream>>>(batch, inv, nscale, GCN_N);

    // ---- CSR build (once, reused for all layers) ----
    gcn_zero_i<<<(GCN_N + 255) / 256, 256, 0, stream>>>(deg, GCN_N);
    gcn_deg<<<(GCN_E + 255) / 256, 256, 0, stream>>>(dst, deg, GCN_E);
    gcn_chunk_reduce<<<nchunks, SCAN_B, 0, stream>>>(deg, chunkSum, GCN_N);
    gcn_chunk_scan<<<1, 32, 0, stream>>>(chunkSum, chunkOff, nchunks);
    gcn_chunk_apply<<<nchunks, SCAN_B, 0, stream>>>(deg, chunkOff, rowptr, GCN_N);
    gcn_copy_i<<<(GCN_N + 255) / 256, 256, 0, stream>>>(rowptr, pos, GCN_N);
    gcn_csr_fill<<<(GCN_E + 255) / 256, 256, 0, stream>>>(src, dst, pos, csr_src, GCN_E);

    // ---- 3 GraphConv layers ----
    const size_t ldsBytes = LDS_TOTAL_FLOATS * sizeof(float);  // 131584 B
    const int gemmBlocks = (NTILES + 7) / 8;                   // 313
    const float* cur = x;
    float* nxt = xA;
    for (int l = 0; l < GCN_L; ++l) {
        gcn_gather<<<(GCN_N * 32 + 255) / 256, 256, 0, stream>>>(cur, rowptr, csr_src, agg, GCN_N);
        gcn_gemm_wmma<<<gemmBlocks, 256, ldsBytes, stream>>>(
            agg, cur,
            w_rel  + (size_t)l * GCN_H * GCN_H,
            w_root + (size_t)l * GCN_H * GCN_H,
            b_rel  + (size_t)l * GCN_H,
            nscale, nxt, NTILES);
        cur = nxt;
        nxt = (nxt == xA) ? xB : xA;
    }

    // ---- mean pool + head ----
    gcn_zero_f4<<<(GCN_G * GCN_H / 4 + 255) / 256, 256, 0, stream>>>(
        (float4*)pooled, GCN_G * GCN_H / 4);
    gcn_pool<<<(GCN_N * 32 + 255) / 256, 256, 0, stream>>>(cur, batch, pooled, GCN_N);
    gcn_head<<<GCN_G, GCN_H, 0, stream>>>(pooled, inv, w1, b1, w2, b2, out);

    (void)in_sizes; (void)n_in; (void)out_size; (void)ws_size;
}